// GINO_Encoder_33071248179438
// MI455X (gfx1250) — compile-verified
//
#include <hip/hip_runtime.h>
#include <math.h>

#define N_IN   100000
#define N_E    262144
#define N_Q    32768
#define PROJ   256
#define HID    80
#define AS     81      // padded LDS activation row stride (floats), odd -> conflict-free A reads
#define WPB    4       // waves per block (128 threads)

typedef float v2f __attribute__((ext_vector_type(2)));
typedef float v8f __attribute__((ext_vector_type(8)));

// CDNA5 f32 WMMA: D(16x16) = A(16x4) * B(4x16) + C, full fp32 (matches fp32 reference).
__device__ __forceinline__ v8f wmma4(v2f a, v2f b, v8f c) {
  return __builtin_amdgcn_wmma_f32_16x16x4_f32(false, a, false, b, (short)0, c, false, false);
}

// Wave-local LDS fence: same-wave DS ops execute in order in HW; the asm blocks
// compiler reordering and drains DScnt.
#define WAVE_SYNC() asm volatile("s_wait_dscnt 0" ::: "memory")

// XLA/Eigen f32 erf: x*P(x^2)/Q(x^2), clamp |x|<=4. This is the exact polynomial
// jax.nn.gelu(approximate=False) lowers to for f32, and it is branch-free
// (vs libm erff's two-interval exec-mask dance).
__device__ __forceinline__ float erf_xla(float x) {
  x = fminf(fmaxf(x, -4.0f), 4.0f);
  const float x2 = x * x;
  float p = -2.72614225801306e-10f;
  p = fmaf(p, x2, 2.77068142495902e-08f);
  p = fmaf(p, x2, -2.10102402082508e-06f);
  p = fmaf(p, x2, -5.69250639462346e-05f);
  p = fmaf(p, x2, -7.34990630326855e-04f);
  p = fmaf(p, x2, -2.95459980854025e-03f);
  p = fmaf(p, x2, -1.60960333262415e-02f);
  p = x * p;
  float q = -1.45660718464996e-05f;
  q = fmaf(q, x2, -2.13374055278905e-04f);
  q = fmaf(q, x2, -1.68282697438203e-03f);
  q = fmaf(q, x2, -7.37332916720468e-03f);
  q = fmaf(q, x2, -1.42647390514189e-02f);
  return p / q;
}

__device__ __forceinline__ float gelu_exact(float v) {
  return 0.5f * v * (1.0f + erf_xla(v * 0.70710678118654752440f));
}

__global__ void __launch_bounds__(256) gino_zero_kernel(float* __restrict__ p, int n) {
  for (int i = blockIdx.x * blockDim.x + threadIdx.x; i < n; i += gridDim.x * blockDim.x)
    p[i] = 0.0f;
}

// Repack a (K x N) row-major weight into WMMA-B operand order:
//   P[((kk*2 + h)*N + n)*2 + v] = (k < K) ? W[k*N + n] : 0,  k = 4*kk + 2*h + v
// so the hot loop fetches each lane's B pair with ONE global_load_b64.
__global__ void __launch_bounds__(256) gino_pack_kernel(const float* __restrict__ W,
                                                        float* __restrict__ P,
                                                        int K, int N) {
  const int nChunks = (K + 3) / 4;
  const int total = nChunks * 2 * N * 2;
  for (int i = blockIdx.x * blockDim.x + threadIdx.x; i < total;
       i += gridDim.x * blockDim.x) {
    const int v = i & 1;
    const int n = (i >> 1) % N;
    const int t = (i >> 1) / N;            // t = kk*2 + h
    const int k = (t >> 1) * 4 + (t & 1) * 2 + v;
    P[i] = (k < K) ? W[k * N + n] : 0.0f;
  }
}

__global__ void __launch_bounds__(128) gino_edge_kernel(
    const float* __restrict__ x,     // (N_IN,3)
    const float* __restrict__ geom,  // (N_IN,3)
    const float* __restrict__ lq,    // (N_Q,3)
    const int*   __restrict__ nidx,  // (N_E)
    const int*   __restrict__ qidx,  // (N_E) sorted
    const float* __restrict__ Wp, const float* __restrict__ bp,   // (3,256),(256)
    const float2* __restrict__ P1, const float* __restrict__ b1,  // packed (8x80)
    const float2* __restrict__ P2, const float* __restrict__ b2,  // packed (80x80)
    const float2* __restrict__ P3, const float* __restrict__ b3,  // packed (80x80)
    const float2* __restrict__ P4, const float* __restrict__ b4,  // packed (80x256)
    float* __restrict__ s, float* __restrict__ cnt)
{
  __shared__ float act[WPB][32 * AS];   // per-wave 32x80 activation tile (two 16-row M-tiles)
  __shared__ float xbuf[WPB][32][3];    // per-wave x[nbr] rows for f recompute
  __shared__ int   qbuf[WPB][32];       // per-wave query ids

  const int lane = threadIdx.x & 31;
  const int wave = threadIdx.x >> 5;
  const int m16  = lane & 15;           // N (or M for A loads) within 16
  const int h    = lane >> 4;           // lane half selects K pair / M+8 rows
  float* wact = act[wave];

  const int nPairs = N_E / 32;          // 8192, exact
  const int wgl = blockIdx.x * WPB + wave;
  const int nw  = gridDim.x * WPB;

  for (int pair = wgl; pair < nPairs; pair += nw) {
    // ---- stage 32 edges (all lanes): h0 = [geom[nbr], lq[q]] (pad K 6->8), x[nbr], q ----
    const int e  = pair * 32 + lane;
    const int nb = nidx[e];
    const int q  = qidx[e];
    qbuf[wave][lane] = q;
    {
      float* r = &wact[lane * AS];
      r[0] = geom[nb * 3 + 0];
      r[1] = geom[nb * 3 + 1];
      r[2] = geom[nb * 3 + 2];
      r[3] = lq[q * 3 + 0];
      r[4] = lq[q * 3 + 1];
      r[5] = lq[q * 3 + 2];
      r[6] = 0.0f;                      // K padding (packed W1 rows 6,7 are zero too)
      r[7] = 0.0f;
      xbuf[wave][lane][0] = x[nb * 3 + 0];
      xbuf[wave][lane][1] = x[nb * 3 + 1];
      xbuf[wave][lane][2] = x[nb * 3 + 2];
    }
    WAVE_SYNC();

    // one count atomic per edge (q still live in a register)
    (void)__hip_atomic_fetch_add(&cnt[q], 1.0f,
                                 __ATOMIC_RELAXED, __HIP_MEMORY_SCOPE_AGENT);

    // ---------------- Layer 1: (32x8) @ (8x80) + b1, GELU ----------------
    {
      // f32 A layout: lane m16 holds M=m16; VGPR v + lane-half h -> K = k0 + 2h + v
      float a0[2][4];
      #pragma unroll
      for (int t = 0; t < 2; ++t) {
        a0[t][0] = wact[(t * 16 + m16) * AS + 2 * h + 0];
        a0[t][1] = wact[(t * 16 + m16) * AS + 2 * h + 1];
        a0[t][2] = wact[(t * 16 + m16) * AS + 4 + 2 * h + 0];
        a0[t][3] = wact[(t * 16 + m16) * AS + 4 + 2 * h + 1];
      }
      WAVE_SYNC();  // A in regs before outputs overwrite the tile
      for (int j = 0; j < 5; ++j) {
        const int n = j * 16 + m16;
        const float2* pj = P1 + (h * HID + n);   // float2 units; chunk stride = 2*HID
        const float bia = b1[n];
        v8f c0 = {bia, bia, bia, bia, bia, bia, bia, bia};
        v8f c1 = c0;
        const float2 bv0 = pj[0];
        const float2 bv1 = pj[2 * HID];
        c0 = wmma4((v2f){a0[0][0], a0[0][1]}, (v2f){bv0.x, bv0.y}, c0);
        c1 = wmma4((v2f){a0[1][0], a0[1][1]}, (v2f){bv0.x, bv0.y}, c1);
        c0 = wmma4((v2f){a0[0][2], a0[0][3]}, (v2f){bv1.x, bv1.y}, c0);
        c1 = wmma4((v2f){a0[1][2], a0[1][3]}, (v2f){bv1.x, bv1.y}, c1);
        // D layout: VGPR r, lane -> (M = r + 8h, N = n)
        #pragma unroll
        for (int r = 0; r < 8; ++r) {
          wact[(r + 8 * h) * AS + n]      = gelu_exact(c0[r]);
          wact[(16 + r + 8 * h) * AS + n] = gelu_exact(c1[r]);
        }
      }
    }
    WAVE_SYNC();

    // ---------------- Layers 2,3: (32x80) @ (80x80) + b, GELU ----------------
    #pragma unroll 1
    for (int L = 0; L < 2; ++L) {
      const float2* P  = (L == 0) ? P2 : P3;
      const float*  bb = (L == 0) ? b2 : b3;
      float A0[40], A1[40];
      #pragma unroll
      for (int kk = 0; kk < 20; ++kk) {
        A0[2 * kk + 0] = wact[m16 * AS + 4 * kk + 2 * h + 0];
        A0[2 * kk + 1] = wact[m16 * AS + 4 * kk + 2 * h + 1];
        A1[2 * kk + 0] = wact[(16 + m16) * AS + 4 * kk + 2 * h + 0];
        A1[2 * kk + 1] = wact[(16 + m16) * AS + 4 * kk + 2 * h + 1];
      }
      WAVE_SYNC();
      for (int j = 0; j < 5; ++j) {
        const int n = j * 16 + m16;
        const float2* pj = P + (h * HID + n);
        const float bia = bb[n];
        v8f c0 = {bia, bia, bia, bia, bia, bia, bia, bia};
        v8f c1 = c0;
        #pragma unroll
        for (int kk = 0; kk < 20; ++kk) {
          const float2 bv = pj[kk * 2 * HID];   // one b64 load, two WMMAs
          c0 = wmma4((v2f){A0[2 * kk], A0[2 * kk + 1]}, (v2f){bv.x, bv.y}, c0);
          c1 = wmma4((v2f){A1[2 * kk], A1[2 * kk + 1]}, (v2f){bv.x, bv.y}, c1);
        }
        #pragma unroll
        for (int r = 0; r < 8; ++r) {
          wact[(r + 8 * h) * AS + n]      = gelu_exact(c0[r]);
          wact[(16 + r + 8 * h) * AS + n] = gelu_exact(c1[r]);
        }
      }
      WAVE_SYNC();
    }

    // ------- Layer 4: (32x80) @ (80x256) + b4, fused f-recompute * scatter -------
    {
      float A0[40], A1[40];
      #pragma unroll
      for (int kk = 0; kk < 20; ++kk) {
        A0[2 * kk + 0] = wact[m16 * AS + 4 * kk + 2 * h + 0];
        A0[2 * kk + 1] = wact[m16 * AS + 4 * kk + 2 * h + 1];
        A1[2 * kk + 0] = wact[(16 + m16) * AS + 4 * kk + 2 * h + 0];
        A1[2 * kk + 1] = wact[(16 + m16) * AS + 4 * kk + 2 * h + 1];
      }
      float xr0[2][8], xr1[2][8], xr2[2][8];
      int qr[2][8];
      #pragma unroll
      for (int t = 0; t < 2; ++t) {
        #pragma unroll
        for (int r = 0; r < 8; ++r) {   // this lane owns rows M = t*16 + 8h + r of D
          const int row = t * 16 + 8 * h + r;
          xr0[t][r] = xbuf[wave][row][0];
          xr1[t][r] = xbuf[wave][row][1];
          xr2[t][r] = xbuf[wave][row][2];
          qr[t][r]  = qbuf[wave][row];
        }
      }
      WAVE_SYNC();
      for (int j = 0; j < 16; ++j) {
        const int n = j * 16 + m16;
        const float2* pj = P4 + (h * PROJ + n);
        const float bia = b4[n];
        v8f c0 = {bia, bia, bia, bia, bia, bia, bia, bia};
        v8f c1 = c0;
        #pragma unroll
        for (int kk = 0; kk < 20; ++kk) {
          const float2 bv = pj[kk * 2 * PROJ];
          c0 = wmma4((v2f){A0[2 * kk], A0[2 * kk + 1]}, (v2f){bv.x, bv.y}, c0);
          c1 = wmma4((v2f){A1[2 * kk], A1[2 * kk + 1]}, (v2f){bv.x, bv.y}, c1);
        }
        // f[nbr][n] = bp[n] + x . W_proj column n (rank-3: 3 FMAs per element)
        const float wp0 = Wp[n], wp1 = Wp[PROJ + n], wp2 = Wp[2 * PROJ + n];
        const float bpp = bp[n];
        #pragma unroll
        for (int t = 0; t < 2; ++t) {
          const v8f& c = (t == 0) ? c0 : c1;
          // run-length segmented reduce over sorted q before atomics
          float accv = 0.0f;
          int curq = qr[t][0];
          #pragma unroll
          for (int r = 0; r < 8; ++r) {
            const float fv = bpp + xr0[t][r] * wp0 + xr1[t][r] * wp1 + xr2[t][r] * wp2;
            const float v  = c[r] * fv;
            const int qq = qr[t][r];
            if (qq != curq) {
              (void)__hip_atomic_fetch_add(&s[curq * PROJ + n], accv,
                                           __ATOMIC_RELAXED, __HIP_MEMORY_SCOPE_AGENT);
              curq = qq;
              accv = v;
            } else {
              accv += v;
            }
          }
          (void)__hip_atomic_fetch_add(&s[curq * PROJ + n], accv,
                                       __ATOMIC_RELAXED, __HIP_MEMORY_SCOPE_AGENT);
        }
      }
    }
  }
}

__global__ void __launch_bounds__(256) gino_final_kernel(const float* __restrict__ s,
                                                         const float* __restrict__ cnt,
                                                         float* __restrict__ out) {
  const int i = blockIdx.x * blockDim.x + threadIdx.x;   // float4 index
  if (i >= (N_Q * PROJ) / 4) return;
  const float4 v = reinterpret_cast<const float4*>(s)[i];
  const float d = fmaxf(cnt[i >> 6], 1.0f);
  float4 o;
  o.x = v.x / d; o.y = v.y / d; o.z = v.z / d; o.w = v.w / d;
  reinterpret_cast<float4*>(out)[i] = o;
}

extern "C" void kernel_launch(void* const* d_in, const int* in_sizes, int n_in,
                              void* d_out, int out_size, void* d_ws, size_t ws_size,
                              hipStream_t stream) {
  const float* x    = (const float*)d_in[0];
  const float* geom = (const float*)d_in[1];
  const float* lq   = (const float*)d_in[2];
  const int*   nidx = (const int*)d_in[3];
  const int*   qidx = (const int*)d_in[4];
  const float* Wp = (const float*)d_in[5];
  const float* bp = (const float*)d_in[6];
  const float* W1 = (const float*)d_in[7];
  const float* b1 = (const float*)d_in[8];
  const float* W2 = (const float*)d_in[9];
  const float* b2 = (const float*)d_in[10];
  const float* W3 = (const float*)d_in[11];
  const float* b3 = (const float*)d_in[12];
  const float* W4 = (const float*)d_in[13];
  const float* b4 = (const float*)d_in[14];

  float* s   = (float*)d_ws;                 // (N_Q, 256) accum
  float* cnt = s + (size_t)N_Q * PROJ;       // (N_Q) edge counts
  float* p1  = cnt + N_Q;                    // packed W1: 2*2*80*2   = 640
  float* p2  = p1 + 640;                     // packed W2: 20*2*80*2  = 6400
  float* p3  = p2 + 6400;                    // packed W3: 20*2*80*2  = 6400
  float* p4  = p3 + 6400;                    // packed W4: 20*2*256*2 = 20480
  float* out = (float*)d_out;

  const int nzero = N_Q * PROJ + N_Q;
  gino_zero_kernel<<<1024, 256, 0, stream>>>(s, nzero);
  gino_pack_kernel<<<8,  256, 0, stream>>>(W1, p1, 6,  HID);
  gino_pack_kernel<<<32, 256, 0, stream>>>(W2, p2, HID, HID);
  gino_pack_kernel<<<32, 256, 0, stream>>>(W3, p3, HID, HID);
  gino_pack_kernel<<<96, 256, 0, stream>>>(W4, p4, HID, PROJ);
  gino_edge_kernel<<<2048, 128, 0, stream>>>(x, geom, lq, nidx, qidx, Wp, bp,
                                             (const float2*)p1, b1, (const float2*)p2, b2,
                                             (const float2*)p3, b3, (const float2*)p4, b4,
                                             s, cnt);
  gino_final_kernel<<<((N_Q * PROJ / 4) + 255) / 256, 256, 0, stream>>>(s, cnt, out);
}